// GraphConv_23476291240269
// MI455X (gfx1250) — compile-verified
//
#include <hip/hip_runtime.h>

typedef float v2f __attribute__((ext_vector_type(2)));
typedef float v4f __attribute__((ext_vector_type(4)));
typedef float v8f __attribute__((ext_vector_type(8)));
typedef int   v4i __attribute__((ext_vector_type(4)));

#define BATCH 8
#define PPTS  16384
#define KQ    16384
#define NNB   16
#define FIN   64
#define NREL  3
#define FOUT  64
#define CIN   67      // IN + REL
#define CPAD  68      // padded to 17 * 4 (K-steps of f32 WMMA)
#define WPB   8       // waves per block
#define RPW   16      // rows per wave

__global__ __launch_bounds__(256) void graphconv_wmma_kernel(
    const float* __restrict__ feats,   // [B, P, 64]   (L2-resident, temporal)
    const int*   __restrict__ n_idxs,  // [B, K, 16]   (streamed, NT)
    const float* __restrict__ nrel,    // [B, K, 16, 3](streamed, NT)
    const int*   __restrict__ nvalid,  // [B, K, 16]   (streamed, NT)
    const float* __restrict__ Wm,      // [64, 67]
    const float* __restrict__ bias,    // [64]
    float*       __restrict__ out)     // [B, K, 64]   (streamed, NT)
{
    __shared__ float ldsA[WPB][RPW][CPAD];   // 34816 B : per-wave A tiles (scaled sums)
    __shared__ float ldsW[FOUT][CPAD];       // 17408 B : padded weights

    const int tid = threadIdx.x;

    // ---- stage padded W into LDS: 256 threads = 64 rows x 4 quarters of 17 ----
    {
        const int o = tid >> 2;          // weight row
        const int q = tid & 3;           // quarter
#pragma unroll
        for (int j = 0; j < 17; ++j) {
            const int c = q * 17 + j;    // 0..67
            ldsW[o][c] = (c < CIN) ? Wm[o * CIN + c] : 0.0f;
        }
    }

    const int wave = tid >> 5;
    const int lane = tid & 31;
    const int lo   = lane & 15;   // row within tile (agg) / M or N index (wmma)
    const int hi   = lane >> 4;   // channel half (agg) / K sub-index (wmma)

    const long long g0 = ((long long)blockIdx.x * WPB + wave) * RPW; // first global row
    const int bb = (int)(g0 / KQ);                                   // batch (tiles never straddle)

    // ---- per-row streamed metadata: idx[16], valid[16], rel[16][3] (NT loads) ----
    const long long grow = g0 + lo;
    const long long nb   = grow * NNB;

    int   idx16[NNB];
    float msk16[NNB];
#pragma unroll
    for (int q = 0; q < 4; ++q) {
        const v4i ii = __builtin_nontemporal_load((const v4i*)(n_idxs + nb) + q);
        const v4i vv = __builtin_nontemporal_load((const v4i*)(nvalid + nb) + q);
        idx16[4 * q + 0] = ii.x; idx16[4 * q + 1] = ii.y;
        idx16[4 * q + 2] = ii.z; idx16[4 * q + 3] = ii.w;
        msk16[4 * q + 0] = (float)vv.x; msk16[4 * q + 1] = (float)vv.y;
        msk16[4 * q + 2] = (float)vv.z; msk16[4 * q + 3] = (float)vv.w;
    }

    float rel48[NNB * NREL];              // only h=1 lanes consume these
    if (hi) {
#pragma unroll
        for (int q = 0; q < 12; ++q) {    // 48 floats = 12 x b128, 16B aligned (grow*192B)
            const v4f rv = __builtin_nontemporal_load((const v4f*)(nrel + nb * NREL) + q);
            rel48[4 * q + 0] = rv.x; rel48[4 * q + 1] = rv.y;
            rel48[4 * q + 2] = rv.z; rel48[4 * q + 3] = rv.w;
        }
    }

    // ---- aggregation: lane (lo,hi) accumulates channels [h*32, h*32+32) + rel (h=1) ----
    float acc[35];
#pragma unroll
    for (int j = 0; j < 35; ++j) acc[j] = 0.0f;
    float cnt = 0.0f;

    const long long fhalf = (long long)bb * PPTS * FIN + hi * 32;
#pragma unroll
    for (int n = 0; n < NNB; ++n) {
        const float m = msk16[n];
        cnt += m;
        const v4f* f4 = (const v4f*)(feats + fhalf + (long long)idx16[n] * FIN);
#pragma unroll
        for (int q = 0; q < 8; ++q) {     // 8 x global_load_b128 (L2 hits)
            const v4f v = f4[q];
            acc[4 * q + 0] += m * v.x;
            acc[4 * q + 1] += m * v.y;
            acc[4 * q + 2] += m * v.z;
            acc[4 * q + 3] += m * v.w;
        }
        if (hi) {
            acc[32] += m * rel48[n * NREL + 0];
            acc[33] += m * rel48[n * NREL + 1];
            acc[34] += m * rel48[n * NREL + 2];
        }
    }

    const float inv = (cnt > 0.0f) ? (1.0f / cnt) : 0.0f;

    // ---- stage A tile into LDS (b128 stores; rows are 272B so 16B-aligned) ----
    {
        v4f* dst = (v4f*)&ldsA[wave][lo][hi * 32];
#pragma unroll
        for (int q = 0; q < 8; ++q) {
            v4f s;
            s.x = acc[4 * q + 0] * inv; s.y = acc[4 * q + 1] * inv;
            s.z = acc[4 * q + 2] * inv; s.w = acc[4 * q + 3] * inv;
            dst[q] = s;
        }
        if (hi) {                          // channels 64..66 + zero pad (67)
            v4f s;
            s.x = acc[32] * inv; s.y = acc[33] * inv;
            s.z = acc[34] * inv; s.w = 0.0f;
            *(v4f*)&ldsA[wave][lo][64] = s;
        }
    }

    __syncthreads();   // A tiles + W staged

    // ---- WMMA: D[16x16] = A[16x68] x W^T[68x16] per column tile ----
    // A frag (f32 16x4): lane l -> M = l&15, VGPR v -> K = v + 2*(l>>4)
    // B frag (f32 4x16): lane l -> N = l&15, VGPR v -> K = v + 2*(l>>4)
#pragma unroll
    for (int t = 0; t < 4; ++t) {
        const int o = t * 16 + lo;        // output channel for this lane's N
        v8f c = {};
#pragma unroll
        for (int kk = 0; kk < 17; ++kk) {
            const int cbase = kk * 4 + 2 * hi;
            v2f a, bf;
            a.x  = ldsA[wave][lo][cbase];
            a.y  = ldsA[wave][lo][cbase + 1];
            bf.x = ldsW[o][cbase];
            bf.y = ldsW[o][cbase + 1];
            c = __builtin_amdgcn_wmma_f32_16x16x4_f32(
                    /*neg_a=*/false, a, /*neg_b=*/false, bf,
                    /*c_mod=*/(short)0, c, /*reuse_a=*/false, /*reuse_b=*/false);
        }
        const float bo = bias[o];
#pragma unroll
        for (int v = 0; v < 8; ++v) {
            const int M = v + 8 * hi;     // D layout: VGPR v, lanes16-31 -> M+8
            const float r = c[v] + bo;
            __builtin_nontemporal_store((r > 0.0f) ? r : 0.0f,
                                        &out[(g0 + M) * FOUT + o]);
        }
    }
}

extern "C" void kernel_launch(void* const* d_in, const int* in_sizes, int n_in,
                              void* d_out, int out_size, void* d_ws, size_t ws_size,
                              hipStream_t stream) {
    // setup_inputs order: keys(0), points(1), feats(2), n_idxs(3),
    //                     neighbor_rel(4), neighbor_valid(5), W(6), b(7)
    const float* feats  = (const float*)d_in[2];
    const int*   n_idxs = (const int*)  d_in[3];
    const float* nrel   = (const float*)d_in[4];
    const int*   nvalid = (const int*)  d_in[5];
    const float* Wm     = (const float*)d_in[6];
    const float* bias   = (const float*)d_in[7];
    float* out = (float*)d_out;

    const long long rows = (long long)BATCH * KQ;          // 131072
    const int blocks = (int)(rows / (WPB * RPW));          // 1024
    hipLaunchKernelGGL(graphconv_wmma_kernel, dim3(blocks), dim3(256), 0, stream,
                       feats, n_idxs, nrel, nvalid, Wm, bias, out);
}